// GCNModel_68023692034522
// MI455X (gfx1250) — compile-verified
//
#include <hip/hip_runtime.h>
#include <math.h>

typedef float v2f __attribute__((ext_vector_type(2)));
typedef float v8f __attribute__((ext_vector_type(8)));

#define DIN 128
#define DH  128
#define DOUT 16

__global__ void k_init(float* __restrict__ agg1, float* __restrict__ agg2,
                       float* __restrict__ deg, int n) {
    long long i = (long long)blockIdx.x * blockDim.x + threadIdx.x;
    long long tot = (long long)n * DH;
    if (i < tot) agg1[i] = 0.0f;
    if (i < (long long)n * DOUT) agg2[i] = 0.0f;
    if (i < n) deg[i] = 1.0f;
}

__global__ void k_degree(const long long* __restrict__ ei, float* __restrict__ deg,
                         int E) {
    int e = blockIdx.x * blockDim.x + threadIdx.x;
    if (e < E) {
        int d = (int)ei[(long long)E + e];
        atomicAdd(&deg[d], 1.0f);
    }
}

__global__ void k_rsqrt(float* __restrict__ deg, int n) {
    int i = blockIdx.x * blockDim.x + threadIdx.x;
    if (i < n) {
        float v = deg[i];
        deg[i] = (v > 0.0f) ? rsqrtf(v) : 0.0f;
    }
}

// GEMM1: C[n x 128] = A[n x 128] * B[128 x 128] via V_WMMA_F32_16X16X4_F32.
__global__ void k_gemm1(const float* __restrict__ A, const float* __restrict__ B,
                        float* __restrict__ C, int n) {
    __shared__ float As[16 * DIN];
    const int r0 = blockIdx.x * 16;
    const int t  = threadIdx.x;
    const bool full = (r0 + 16 <= n);   // uniform across the block

    {
        int row = t >> 4;
        int cb  = (t & 15) * 8;
        int gr  = r0 + row;
        if (!full && gr > n - 1) gr = n - 1;
        const float4* src = (const float4*)(A + (size_t)gr * DIN + cb);
        float4 v0 = src[0];
        float4 v1 = src[1];
        *(float4*)(As + row * DIN + cb)     = v0;
        *(float4*)(As + row * DIN + cb + 4) = v1;
    }
    __syncthreads();

    const int wave = t >> 5;
    const int lane = t & 31;
    const int half = lane >> 4;
    const int l    = lane & 15;
    const int c0   = wave * 16;

    v8f acc = {};
    #pragma unroll 4
    for (int k0 = 0; k0 < DIN; k0 += 4) {
        const int ka = k0 + 2 * half;
        v2f a, b;
        a.x = As[l * DIN + ka];
        a.y = As[l * DIN + ka + 1];
        b.x = B[(size_t)ka * DH + c0 + l];
        b.y = B[(size_t)(ka + 1) * DH + c0 + l];
        acc = __builtin_amdgcn_wmma_f32_16x16x4_f32(
            false, a, false, b, (short)0, acc, false, false);
    }

    float* cptr = C + (size_t)(r0 + 8 * half) * DH + c0 + l;
    if (full) {
        #pragma unroll
        for (int j = 0; j < 8; ++j) cptr[(size_t)j * DH] = acc[j];
    } else {
        #pragma unroll
        for (int j = 0; j < 8; ++j) {
            if (r0 + j + 8 * half < n) cptr[(size_t)j * DH] = acc[j];
        }
    }
}

__global__ void k_scatter1(const long long* __restrict__ ei,
                           const float* __restrict__ dinv,
                           const float* __restrict__ t0,
                           float* __restrict__ agg1, int E, int n) {
    const int lane = threadIdx.x & 31;
    long long e = (long long)blockIdx.x * 8 + (threadIdx.x >> 5);
    const long long EA = (long long)E + n;
    if (e >= EA) return;
    long long s, d;
    if (e < E) { s = ei[e]; d = ei[(long long)E + e]; }
    else       { s = d = e - E; }
    const float coef = dinv[s] * dinv[d];
    const float4 v = *(const float4*)(t0 + s * DH + lane * 4);
    float* dst = agg1 + d * DH + lane * 4;
    atomicAdd(dst + 0, v.x * coef);
    atomicAdd(dst + 1, v.y * coef);
    atomicAdd(dst + 2, v.z * coef);
    atomicAdd(dst + 3, v.w * coef);
}

__global__ void k_bias_relu(float* __restrict__ h, const float* __restrict__ b1,
                            long long total) {
    long long i = (long long)blockIdx.x * blockDim.x + threadIdx.x;
    if (i < total) {
        int j = (int)(i & (DH - 1));
        h[i] = fmaxf(h[i] + b1[j], 0.0f);
    }
}

// GEMM2: t2[n x 16] = h[n x 128] * W2[128 x 16].  One wave per 16-row tile.
__global__ void k_gemm2(const float* __restrict__ A, const float* __restrict__ B,
                        float* __restrict__ C, int n) {
    const int r0   = blockIdx.x * 16;
    const int lane = threadIdx.x & 31;
    const int half = lane >> 4;
    const int l    = lane & 15;
    const bool full = (r0 + 16 <= n);
    int ra = r0 + l;
    if (!full && ra > n - 1) ra = n - 1;

    v8f acc = {};
    #pragma unroll 4
    for (int k0 = 0; k0 < DH; k0 += 4) {
        const int ka = k0 + 2 * half;
        v2f a, b;
        a.x = A[(size_t)ra * DH + ka];
        a.y = A[(size_t)ra * DH + ka + 1];
        b.x = B[ka * DOUT + l];
        b.y = B[(ka + 1) * DOUT + l];
        acc = __builtin_amdgcn_wmma_f32_16x16x4_f32(
            false, a, false, b, (short)0, acc, false, false);
    }

    float* cptr = C + (size_t)(r0 + 8 * half) * DOUT + l;
    if (full) {
        #pragma unroll
        for (int j = 0; j < 8; ++j) cptr[(size_t)j * DOUT] = acc[j];
    } else {
        #pragma unroll
        for (int j = 0; j < 8; ++j) {
            if (r0 + j + 8 * half < n) cptr[(size_t)j * DOUT] = acc[j];
        }
    }
}

__global__ void k_scatter2(const long long* __restrict__ ei,
                           const float* __restrict__ dinv,
                           const float* __restrict__ t2,
                           float* __restrict__ agg2, int E, int n) {
    long long idx = (long long)blockIdx.x * blockDim.x + threadIdx.x;
    const long long EA = (long long)E + n;
    if (idx >= EA * DOUT) return;
    long long e = idx >> 4;
    const int c = (int)(idx & (DOUT - 1));
    long long s, d;
    if (e < E) { s = ei[e]; d = ei[(long long)E + e]; }
    else       { s = d = e - E; }
    const float coef = dinv[s] * dinv[d];
    atomicAdd(agg2 + d * DOUT + c, t2[s * DOUT + c] * coef);
}

__global__ void k_lsm(const float* __restrict__ agg2, const float* __restrict__ b2,
                      float* __restrict__ out, int n) {
    int i = blockIdx.x * blockDim.x + threadIdx.x;
    if (i >= n) return;
    float v[DOUT];
    float mx = -INFINITY;
    #pragma unroll
    for (int c = 0; c < DOUT; ++c) {
        v[c] = agg2[(size_t)i * DOUT + c] + b2[c];
        mx = fmaxf(mx, v[c]);
    }
    float sum = 0.0f;
    #pragma unroll
    for (int c = 0; c < DOUT; ++c) sum += expf(v[c] - mx);
    const float ls = logf(sum) + mx;
    #pragma unroll
    for (int c = 0; c < DOUT; ++c) out[(size_t)i * DOUT + c] = v[c] - ls;
}

extern "C" void kernel_launch(void* const* d_in, const int* in_sizes, int n_in,
                              void* d_out, int out_size, void* d_ws, size_t ws_size,
                              hipStream_t stream) {
    const float*     x  = (const float*)d_in[0];
    const long long* ei = (const long long*)d_in[1];
    const float*     W1 = (const float*)d_in[2];
    const float*     b1 = (const float*)d_in[3];
    const float*     W2 = (const float*)d_in[4];
    const float*     b2 = (const float*)d_in[5];
    float*           out = (float*)d_out;

    const int n = in_sizes[0] / DIN;   // 100000
    const int E = in_sizes[1] / 2;     // 1600000
    const long long EA = (long long)E + n;

    char* ws = (char*)d_ws;
    size_t off = 0;
    auto slab = [&](size_t bytes) -> void* {
        void* p = ws + off;
        off += (bytes + 255) & ~(size_t)255;
        return p;
    };
    float* dinv = (float*)slab((size_t)n * sizeof(float));
    float* t0   = (float*)slab((size_t)n * DH * sizeof(float));
    float* agg1 = (float*)slab((size_t)n * DH * sizeof(float));
    float* t2   = (float*)slab((size_t)n * DOUT * sizeof(float));
    float* agg2 = (float*)slab((size_t)n * DOUT * sizeof(float));

    const long long nDH = (long long)n * DH;

    k_init<<<(unsigned)((nDH + 255) / 256), 256, 0, stream>>>(agg1, agg2, dinv, n);
    k_degree<<<(E + 255) / 256, 256, 0, stream>>>(ei, dinv, E);
    k_rsqrt<<<(n + 255) / 256, 256, 0, stream>>>(dinv, n);
    k_gemm1<<<(n + 15) / 16, 256, 0, stream>>>(x, W1, t0, n);
    k_scatter1<<<(unsigned)((EA + 7) / 8), 256, 0, stream>>>(ei, dinv, t0, agg1, E, n);
    k_bias_relu<<<(unsigned)((nDH + 255) / 256), 256, 0, stream>>>(agg1, b1, nDH);
    k_gemm2<<<(n + 15) / 16, 32, 0, stream>>>(agg1, W2, t2, n);
    k_scatter2<<<(unsigned)((EA * DOUT + 255) / 256), 256, 0, stream>>>(ei, dinv, t2, agg2, E, n);
    k_lsm<<<(n + 255) / 256, 256, 0, stream>>>(agg2, b2, out, n);
}